// SpatialLoss_27453430956391
// MI455X (gfx1250) — compile-verified
//
#include <hip/hip_runtime.h>
#include <hip/hip_bf16.h>

typedef __attribute__((ext_vector_type(2))) float v2f;
typedef __attribute__((ext_vector_type(8))) float v8f;

#define NSEG   2048
#define C_TOT  64
#define CHB    16                 // channels per workgroup (LDS tile)
#define NCB    (C_TOT / CHB)      // 4 channel blocks
#define PIX    (1024 * 1024)      // pixels in batch 0
#define TPB    256
#define PBLK   64                 // pixel blocks per channel block
#define CHUNK  (PIX / PBLK)       // 16384 pixels per workgroup

// Workspace layout (floats):
//   [0, NSEG*C_TOT)        seg_sum[s*64 + c]
//   [+NSEG)                total_sumsq[s]  (sumsq summed over ALL channels)
//   [+NSEG)                counts[s]
#define WS_FLOATS (NSEG * C_TOT + 2 * NSEG)

#define GATOMIC(p, v) __hip_atomic_fetch_add((p), (v), __ATOMIC_RELAXED, __HIP_MEMORY_SCOPE_AGENT)
#define LATOMIC(p, v) __hip_atomic_fetch_add((p), (v), __ATOMIC_RELAXED, __HIP_MEMORY_SCOPE_WORKGROUP)

__global__ __launch_bounds__(TPB) void zero_ws_kernel(float* __restrict__ ws) {
  int stride = gridDim.x * blockDim.x;
  for (int i = blockIdx.x * blockDim.x + threadIdx.x; i < WS_FLOATS; i += stride)
    ws[i] = 0.0f;
}

// One streaming pass over feats[0]. LDS-privatized scatter-reduce, 16 channels
// per workgroup (128KB sum tile + 8KB sumsq + 8KB counts = 144KB dynamic LDS).
__global__ __launch_bounds__(TPB) void accumulate_kernel(
    const int* __restrict__ sp, const float* __restrict__ feats,
    float* __restrict__ ws) {
  extern __shared__ float lds[];
  float* s_sum = lds;                    // [NSEG*CHB]
  float* s_ssq = lds + NSEG * CHB;       // [NSEG]
  float* s_cnt = s_ssq + NSEG;           // [NSEG]

  float* g_sum = ws;
  float* g_ssq = ws + NSEG * C_TOT;
  float* g_cnt = g_ssq + NSEG;

  const int tid = threadIdx.x;
  const int pb  = blockIdx.x;
  const int cb  = blockIdx.y;

  for (int i = tid; i < NSEG * CHB + 2 * NSEG; i += TPB) lds[i] = 0.0f;
  __syncthreads();

  const int p0 = pb * CHUNK;
  const float* f = feats + (size_t)cb * CHB * PIX;

  for (int it = 0; it < CHUNK; it += TPB * 4) {
    const int p = p0 + it + tid * 4;
    const int4 s4 = *(const int4*)(sp + p);
    // prefetch next segment-id tile (speculative; dropped if OOB)
    __builtin_prefetch(sp + p + TPB * 4, 0, 0);

    float sq0 = 0.f, sq1 = 0.f, sq2 = 0.f, sq3 = 0.f;
#pragma unroll
    for (int c = 0; c < CHB; ++c) {
      const float4 x = *(const float4*)(f + (size_t)c * PIX + p);
      LATOMIC(&s_sum[s4.x * CHB + c], x.x);
      LATOMIC(&s_sum[s4.y * CHB + c], x.y);
      LATOMIC(&s_sum[s4.z * CHB + c], x.z);
      LATOMIC(&s_sum[s4.w * CHB + c], x.w);
      sq0 += x.x * x.x; sq1 += x.y * x.y;
      sq2 += x.z * x.z; sq3 += x.w * x.w;
    }
    LATOMIC(&s_ssq[s4.x], sq0);
    LATOMIC(&s_ssq[s4.y], sq1);
    LATOMIC(&s_ssq[s4.z], sq2);
    LATOMIC(&s_ssq[s4.w], sq3);
    if (cb == 0) {
      LATOMIC(&s_cnt[s4.x], 1.0f);
      LATOMIC(&s_cnt[s4.y], 1.0f);
      LATOMIC(&s_cnt[s4.z], 1.0f);
      LATOMIC(&s_cnt[s4.w], 1.0f);
    }
  }
  __syncthreads();

  // Flush LDS tile to the global accumulators (all L2-resident: 540KB total)
  for (int i = tid; i < NSEG * CHB; i += TPB) {
    const int s = i / CHB;
    const int c = i % CHB;
    GATOMIC(&g_sum[s * C_TOT + cb * CHB + c], s_sum[i]);
  }
  for (int s = tid; s < NSEG; s += TPB) {
    GATOMIC(&g_ssq[s], s_ssq[s]);
    if (cb == 0) GATOMIC(&g_cnt[s], s_cnt[s]);
  }
}

// Finalize: q[s] = sum_c seg_sum[s][c]^2 via V_WMMA_F32_16X16X4_F32 Gram diag,
// then per_seg -> masked sums -> scalar out. One workgroup (16 waves).
__global__ __launch_bounds__(512) void finalize_kernel(
    const float* __restrict__ ws, float* __restrict__ out) {
  const float* g_sum = ws;
  const float* g_ssq = ws + NSEG * C_TOT;
  const float* g_cnt = g_ssq + NSEG;

  __shared__ float s_loss;
  __shared__ float s_c;
  if (threadIdx.x == 0) { s_loss = 0.0f; s_c = 0.0f; }
  __syncthreads();

  const int lane = threadIdx.x & 31;
  const int wave = threadIdx.x >> 5;
  const int m = lane & 15;      // A-matrix row (M) held by this lane
  const int h = lane >> 4;      // lanes 16-31 hold K = 2,3 of each K-quad

  for (int g = wave; g < NSEG / 16; g += 16) {   // uniform 8 iters per wave
    const int seg_base = g * 16;
    const float* row = g_sum + (size_t)(seg_base + m) * C_TOT + 2 * h;

    v8f acc = {0.f, 0.f, 0.f, 0.f, 0.f, 0.f, 0.f, 0.f};
#pragma unroll
    for (int t = 0; t < 16; ++t) {               // K = 64 in steps of 4
      // lane data for A (16x4) and B (4x16) coincide for S * S^T
      v2f a = *(const v2f*)(row + 4 * t);        // {S[m][4t+2h], S[m][4t+2h+1]}
      acc = __builtin_amdgcn_wmma_f32_16x16x4_f32(
          /*neg_a=*/false, a, /*neg_b=*/false, a,
          /*c_mod=*/(short)0, acc, /*reuse_a=*/false, /*reuse_b=*/false);
    }

    // Diagonal of the 16x16 Gram:
    //  m<8 : VGPR m,   lane m        ;  m>=8 : VGPR m-8, lane m+16
    float tmp[8];
#pragma unroll
    for (int i = 0; i < 8; ++i) tmp[i] = acc[i];

    const bool own = (lane < 8) || (lane >= 24);
    const int idx = ((lane < 8) ? lane : (lane - 24)) & 7;
    const int s   = seg_base + ((lane < 8) ? lane : (lane - 16));
    if (own) {
      const float q    = tmp[idx];
      const float cnt  = g_cnt[s];
      const float n    = fmaxf(cnt, 1.0f);
      const float vsum = g_ssq[s] - q / n;         // sum_c var_sum[s][c]
      const float per  = vsum / (64.0f * n);
      if (cnt >= 2.0f) LATOMIC(&s_loss, per);
      if (cnt >  0.0f) LATOMIC(&s_c, 1.0f);
    }
  }
  __syncthreads();
  if (threadIdx.x == 0) out[0] = s_loss / s_c;
}

extern "C" void kernel_launch(void* const* d_in, const int* in_sizes, int n_in,
                              void* d_out, int out_size, void* d_ws, size_t ws_size,
                              hipStream_t stream) {
  const int*   sp    = (const int*)d_in[0];     // [2,1024,1024] int32, batch 0 used
  const float* feats = (const float*)d_in[1];   // [2,64,1024,1024] f32, batch 0 used
  float*       out   = (float*)d_out;
  float*       ws    = (float*)d_ws;

  (void)in_sizes; (void)n_in; (void)out_size; (void)ws_size;

  zero_ws_kernel<<<132, TPB, 0, stream>>>(ws);

  const size_t shmem = (size_t)(NSEG * CHB + 2 * NSEG) * sizeof(float); // 147456 B
  hipFuncSetAttribute((const void*)accumulate_kernel,
                      hipFuncAttributeMaxDynamicSharedMemorySize, (int)shmem);
  dim3 grid(PBLK, NCB);
  accumulate_kernel<<<grid, TPB, shmem, stream>>>(sp, feats, ws);

  finalize_kernel<<<1, 512, 0, stream>>>(ws, out);
}